// MessagePassingBase_7645041787179
// MI455X (gfx1250) — compile-verified
//
#include <hip/hip_runtime.h>

#define D_FEAT 64

// ---------------------------------------------------------------------------
// Kernel 1: residual init  out[i] = input[i], 16B per lane, fully coalesced.
// d_out is poisoned before timing, so we must fully rewrite it every call.
// ---------------------------------------------------------------------------
__global__ void mp_residual_init(const float4* __restrict__ in,
                                 float4* __restrict__ out,
                                 int n4) {
    int i = blockIdx.x * blockDim.x + threadIdx.x;
    if (i < n4) {
        out[i] = in[i];
    }
}

// ---------------------------------------------------------------------------
// Kernel 2: edge gather-scale-scatter, float4 per thread.
//   thread tid -> edge e = tid>>4, float4 slot f4 = tid&15
//   Lanes 0-15 of a wave cover one full 256B source row (one b128 request),
//   lanes 16-31 the next edge's row. The 4 f32 atomics per thread present
//   contiguous 16-lane address groups per instruction, so the L2 atomic units
//   (output is 25.6MB, fully resident in the 192MB L2) can apply each edge
//   row in a minimal number of line-level RMW transactions.
// ---------------------------------------------------------------------------
__global__ void mp_edge_scatter(const float* __restrict__ x,
                                const float* __restrict__ w,
                                const int* __restrict__ src,
                                const int* __restrict__ dst,
                                float* __restrict__ out,
                                int n_edges) {
    int tid = blockIdx.x * blockDim.x + threadIdx.x;   // max 25.6M, fits i32
    int e  = tid >> 4;
    int f4 = tid & 15;
    if (e >= n_edges) return;

    // One lane per wave prefetches the streaming edge arrays ahead
    // (gfx1250 global_prefetch_b8; speculative -> OOB silently dropped).
    if ((threadIdx.x & 31) == 0) {
        __builtin_prefetch(&src[e + 4096], 0, 1);
        __builtin_prefetch(&dst[e + 4096], 0, 1);
        __builtin_prefetch(&w[e + 4096], 0, 1);
    }

    // Per-edge scalars: 16 lanes share each address -> coalesced to one
    // request per distinct edge (2 per wave).
    int   s  = src[e];
    int   d  = dst[e];
    float ww = w[e];

    // Gather one float4 of the source row (global_load_b128, L2-resident).
    const float4* xrow = (const float4*)(x + s * D_FEAT);
    float4 v = xrow[f4];
    v.x *= ww;  v.y *= ww;  v.z *= ww;  v.w *= ww;

    // Scatter-add: relaxed, agent-scope, dead-return -> native no-return
    // global_atomic_add_f32 (STOREcnt-tracked), no CAS loop.
    float* o = out + d * D_FEAT + (f4 << 2);
    (void)__hip_atomic_fetch_add(o + 0, v.x, __ATOMIC_RELAXED, __HIP_MEMORY_SCOPE_AGENT);
    (void)__hip_atomic_fetch_add(o + 1, v.y, __ATOMIC_RELAXED, __HIP_MEMORY_SCOPE_AGENT);
    (void)__hip_atomic_fetch_add(o + 2, v.z, __ATOMIC_RELAXED, __HIP_MEMORY_SCOPE_AGENT);
    (void)__hip_atomic_fetch_add(o + 3, v.w, __ATOMIC_RELAXED, __HIP_MEMORY_SCOPE_AGENT);
}

// ---------------------------------------------------------------------------
// Launch: inputs in setup_inputs() order:
//   d_in[0] = input       (N_NODES * D_FEAT, float32)
//   d_in[1] = edge_weight (N_EDGES, float32)
//   d_in[2] = edge_src    (N_EDGES, int)
//   d_in[3] = edge_dst    (N_EDGES, int)
// d_out = N_NODES * D_FEAT float32
// ---------------------------------------------------------------------------
extern "C" void kernel_launch(void* const* d_in, const int* in_sizes, int n_in,
                              void* d_out, int out_size, void* d_ws, size_t ws_size,
                              hipStream_t stream) {
    const float* x   = (const float*)d_in[0];
    const float* w   = (const float*)d_in[1];
    const int*   src = (const int*)d_in[2];
    const int*   dst = (const int*)d_in[3];
    float*       out = (float*)d_out;

    const int n_node_elems = in_sizes[0];        // N_NODES * D_FEAT
    const int n_edges      = in_sizes[1];        // N_EDGES

    // Kernel 1: out = input (float4 copies)
    {
        int n4    = n_node_elems / 4;
        int block = 256;
        int grid  = (n4 + block - 1) / block;
        mp_residual_init<<<grid, block, 0, stream>>>(
            (const float4*)x, (float4*)out, n4);
    }

    // Kernel 2: scatter-add edge messages (same stream -> ordered after init)
    {
        long long total = (long long)n_edges * 16;   // 16 threads per edge
        int block = 256;
        int grid  = (int)((total + block - 1) / block);
        mp_edge_scatter<<<grid, block, 0, stream>>>(
            x, w, src, dst, out, n_edges);
    }
}